// local_grad_neo_hookean_26706106646780
// MI455X (gfx1250) — compile-verified
//
#include <hip/hip_runtime.h>
#include <math.h>

#define S  192
#define P  191          // output / diff domain extent per axis
#define TD 4
#define TH 16
#define TW 16

// LDS layout (float offsets). Pitch-19 rows: linear index == LDS offset for YB.
#define YB0_OFF 0                    // y tile buf0: [7][19][19] = 2527
#define YB1_OFF 2527                 // y tile buf1: [7][19][19] = 2527
#define DB_OFF  5054                 // diffs: 3 x [6][18][19] = 6156
#define SH_OFF  11210                // 2D-filtered: 3 x [6][16][16] = 4608
#define LDS_TOTAL 15818              // floats = 63272 bytes (< 64 KB static)

#define YB_ELEMS 2527                // 7*19*19
#define LOADS_PER_THREAD 10          // ceil(2527/256); every wave issues exactly 10

// ---- CDNA5 async global->LDS copy (ASYNCcnt path), via inline asm ----
__device__ __forceinline__ void async_g2l_b32(unsigned lds_byte_addr, const float* gsrc) {
    asm volatile("global_load_async_to_lds_b32 %0, %1, off"
                 :: "v"(lds_byte_addr), "v"(gsrc)
                 : "memory");
}
template <int N>
__device__ __forceinline__ void wait_async_le() {
    asm volatile("s_wait_asynccnt %0" :: "n"(N) : "memory");
}

__device__ __forceinline__ int clamp191(int v) {
    return v < 0 ? 0 : (v > 191 ? 191 : v);
}

// Issue the whole 7x19x19 halo tile of channel c as async B32 copies into LDS.
// Exactly LOADS_PER_THREAD issues per thread (tail clamps to last element:
// duplicate async writes of identical data are benign) so that every wave's
// ASYNCcnt contribution is exactly LOADS_PER_THREAD.
__device__ __forceinline__ void issue_tile(const float* __restrict__ y, int c,
                                           int d0, int h0, int w0,
                                           unsigned lds_base, unsigned yb_off, int t) {
#pragma unroll
    for (int k = 0; k < LOADS_PER_THREAD; ++k) {
        int i = t + k * 256;
        if (i >= YB_ELEMS) i = YB_ELEMS - 1;
        int dd = i / 361, rem = i - dd * 361;
        int hh = rem / 19, ww = rem - hh * 19;
        int gd = clamp191(d0 - 1 + dd);   // clamped entries are never read
        int gh = clamp191(h0 - 1 + hh);
        int gw = clamp191(w0 - 1 + ww);
        const float* src = y + (((size_t)c * S + gd) * S + gh) * S + gw;
        unsigned dst = lds_base + 4u * (unsigned)(yb_off + (unsigned)i);  // pitch==rowlen -> i is the offset
        async_g2l_b32(dst, src);
    }
}

__global__ __launch_bounds__(256)
void jac_loss_kernel(const float* __restrict__ y, double* __restrict__ acc) {
    __shared__ float smem[LDS_TOTAL];
    const int t  = threadIdx.x;
    const int tx = t & 15;          // w within tile
    const int ty = t >> 4;          // h within tile
    const int w0 = blockIdx.x * TW;
    const int h0 = blockIdx.y * TH;
    const int d0 = blockIdx.z * TD;
    // Low 32 bits of a generic LDS pointer == LDS byte offset.
    const unsigned lds_base = (unsigned)(uintptr_t)(void*)smem;

    // F[c][g][d]: g: 0 = D-shift filter (fy), 1 = H-shift (fx), 2 = W-shift (fz)
    float F[3][3][TD];

    // Prefetch channel 0 into buf0.
    issue_tile(y, 0, d0, h0, w0, lds_base, YB0_OFF, t);

    for (int c = 0; c < 3; ++c) {
        const unsigned yb = (c & 1) ? YB1_OFF : YB0_OFF;

        // Prefetch next channel into the other buffer, then wait for *this*
        // channel's 10 per-wave loads only (async loads complete in order).
        if (c == 0) { issue_tile(y, 1, d0, h0, w0, lds_base, YB1_OFF, t); wait_async_le<LOADS_PER_THREAD>(); }
        else if (c == 1) { issue_tile(y, 2, d0, h0, w0, lds_base, YB0_OFF, t); wait_async_le<LOADS_PER_THREAD>(); }
        else { wait_async_le<0>(); }
        __syncthreads();

        // ---- Stage 2: |diff| arrays on the 6x18x18 halo diff tile (zero outside [0,191)^3) ----
        for (int i = t; i < 6 * 18 * 18; i += 256) {
            int dd = i / 324, rem = i - dd * 324;
            int hh = rem / 18, ww = rem - hh * 18;
            int dp = d0 - 1 + dd, hp = h0 - 1 + hh, wp = w0 - 1 + ww;
            float vD = 0.f, vH = 0.f, vW = 0.f;
            if (dp >= 0 && dp < P && hp >= 0 && hp < P && wp >= 0 && wp < P) {
                int yi = (int)yb + (dd * 19 + hh) * 19 + ww;
                float b = smem[yi];
                vD = fabsf(smem[yi + 361] - b);   // +1 in d
                vH = fabsf(smem[yi + 19]  - b);   // +1 in h
                vW = fabsf(smem[yi + 1]   - b);   // +1 in w
            }
            int o = (dd * 18 + hh) * 19 + ww;     // DB pitch 19
            smem[DB_OFF + o]        = vD;
            smem[DB_OFF + 2052 + o] = vH;
            smem[DB_OFF + 4104 + o] = vW;
        }
        __syncthreads();

        // ---- Stage 3 (fused W+H): 9-point 2D sum [6][18][19] -> [6][16][16] ----
        for (int i = t; i < 6 * 16 * 16; i += 256) {
            int dd = i >> 8, rem = i & 255;
            int h = rem >> 4, w = rem & 15;
            int ib = (dd * 18 + h) * 19 + w;
#pragma unroll
            for (int g = 0; g < 3; ++g) {
                const float* db = smem + DB_OFF + g * 2052;
                float s = db[ib]      + db[ib + 1]      + db[ib + 2]
                        + db[ib + 19] + db[ib + 20]     + db[ib + 21]
                        + db[ib + 38] + db[ib + 39]     + db[ib + 40];
                smem[SH_OFF + g * 1536 + (dd * 16 + h) * 16 + w] = s;
            }
        }
        __syncthreads();

        // ---- Stage 4: 3-tap sum along D into registers, /27 ----
        const float inv27 = 1.0f / 27.0f;
#pragma unroll
        for (int d = 0; d < TD; ++d) {
            int ib = (d * 16 + ty) * 16 + tx;
#pragma unroll
            for (int g = 0; g < 3; ++g) {
                const float* sh = smem + SH_OFF + g * 1536;
                F[c][g][d] = (sh[ib] + sh[ib + 256] + sh[ib + 512]) * inv27;
            }
        }
        __syncthreads();   // protect DB/SH (and YB buffers) before next channel
    }

    // ---- Pointwise determinant / trace / loss ----
    float local = 0.f;
    if ((h0 + ty < P) && (w0 + tx < P)) {
#pragma unroll
        for (int d = 0; d < TD; ++d) {
            if (d0 + d < P) {
                float dydy = F[0][0][d], dydx = F[0][1][d], dydz = F[0][2][d];
                float dxdy = F[1][0][d], dxdx = F[1][1][d], dxdz = F[1][2][d];
                float dzdy = F[2][0][d], dzdx = F[2][1][d], dzdz = F[2][2][d];
                float a = dxdx + 1.f, e = dydy + 1.f, ii = dzdz + 1.f;
                float det1 = a * (e * ii - dydz * dzdy);
                float det2 = -dxdy * (dydx * ii - dydz * dzdx);
                float det3 = dxdz * (dydx * dzdy - e * dzdx);
                float J = det1 + det2 + det3;
                float Tr = a * a + dxdy * dxdy + dxdz * dxdz
                         + dydx * dydx + e * e + dydz * dydz
                         + dzdx * dzdx + dzdy * dzdy + ii * ii;
                float stretch = Tr * expf(1.f - J) - 3.f;
                float vol = (J - 1.f) * (J - 1.f);
                local += (1.0f / 12.0f) * stretch + (15.0f / 31.0f) * vol;
            }
        }
    }

    // ---- Block reduction (reuse LDS) + f64 global accumulate ----
    smem[t] = local;
    __syncthreads();
#pragma unroll
    for (int s2 = 128; s2 > 0; s2 >>= 1) {
        if (t < s2) smem[t] += smem[t + s2];
        __syncthreads();
    }
    if (t == 0) atomicAdd(acc, (double)smem[0]);
}

__global__ void finalize_kernel(const double* __restrict__ acc, float* __restrict__ out) {
    out[0] = (float)(acc[0] / ((double)P * (double)P * (double)P));
}

extern "C" void kernel_launch(void* const* d_in, const int* in_sizes, int n_in,
                              void* d_out, int out_size, void* d_ws, size_t ws_size,
                              hipStream_t stream) {
    (void)in_sizes; (void)n_in; (void)out_size; (void)ws_size;
    const float* y = (const float*)d_in[0];   // y_pred [1,3,192,192,192]; y_true unused
    double* acc = (double*)d_ws;
    hipMemsetAsync(acc, 0, sizeof(double), stream);   // graph-capturable memset node
    dim3 grid((P + TW - 1) / TW, (P + TH - 1) / TH, (P + TD - 1) / TD);  // 12 x 12 x 48
    jac_loss_kernel<<<grid, 256, 0, stream>>>(y, acc);
    finalize_kernel<<<1, 1, 0, stream>>>(acc, (float*)d_out);
}